// MaxIntersection_47519518163582
// MI455X (gfx1250) — compile-verified
//
#include <hip/hip_runtime.h>
#include <math.h>

typedef float v2f __attribute__((ext_vector_type(2)));
typedef float v8f __attribute__((ext_vector_type(8)));

#define BB      2
#define NN      6890
#define NFACES  2000
#define NTILES  ((NN + 15) / 16)        /* 431 */
#define BIGF    1e10f
#define FOURPI  12.566370614359172f
#define CHUNK   256

// ---------------------------------------------------------------------------
// Kernel 1: per-row minimum distance via V_WMMA_F32_16X16X4_F32.
// Each wave owns one 16-row tile of X, loops over all 16-col tiles of Y.
// A/B layout (CDNA5 ISA): lane holds row (lane&15); VGPR j holds K = j + 2*(lane>>4).
// K=3 dot padded to K=4 with zero. D layout: VGPR r = row r + 8*(lane>>4), col = lane&15.
// Tracks SQUARED minima in the loop (sqrt is monotone on clamped d^2) so the
// expensive correctly-rounded sqrt runs only 8x per wave, not 8x per tile.
// Y-tile loads are double-buffered to hide L2 latency behind the WMMA.
// ---------------------------------------------------------------------------
__global__ void __launch_bounds__(256)
rowmin_wmma_kernel(const float* __restrict__ X, const float* __restrict__ Y,
                   float* __restrict__ rowmin)
{
    const int wave = (blockIdx.x * blockDim.x + threadIdx.x) >> 5;
    const int lane = threadIdx.x & 31;
    const int half = lane >> 4;
    const int m16  = lane & 15;
    if (wave >= BB * NTILES) return;          // wave-uniform exit, EXEC stays full

    const int b  = wave / NTILES;
    const int rt = wave % NTILES;
    const float* Yb = Y + (size_t)b * NN * 3;

    // ---- A fragment: this tile's 16 rows of X (loop-invariant) ----
    int row  = rt * 16 + m16;
    int rowc = row < NN ? row : NN - 1;       // clamp: duplicate row, never written out
    const float* xr = X + ((size_t)b * NN + rowc) * 3;
    float x0 = xr[0], x1 = xr[1], x2 = xr[2];
    float nx_mine = x0 * x0 + x1 * x1 + x2 * x2;
    v2f afrag;
    afrag.x = half ? x2  : x0;                // K = 0 or 2
    afrag.y = half ? 0.f : x1;                // K = 1 or 3 (pad)

    // broadcast |x_m|^2 for the 8 rows this lane's D fragment covers
    float nx[8];
#pragma unroll
    for (int r = 0; r < 8; ++r)
        nx[r] = __shfl(nx_mine, r + 8 * half, 32);

    float rm[8];                               // running SQUARED minima
#pragma unroll
    for (int r = 0; r < 8; ++r) rm[r] = BIGF;

    // ---- double-buffered column-tile loop ----
    int col0 = m16;                            // tile 0 column (always < NN)
    float cy0 = Yb[col0 * 3 + 0], cy1 = Yb[col0 * 3 + 1], cy2 = Yb[col0 * 3 + 2];

    for (int ct = 0; ct < NTILES; ++ct) {
        // issue next tile's load first (clamped; last iteration re-loads harmlessly)
        int ctn  = ct + 1 < NTILES ? ct + 1 : NTILES - 1;
        int coln = ctn * 16 + m16;
        int colnc = coln < NN ? coln : NN - 1; // clamp: duplicates a real point, min unaffected
        const float* yn = Yb + (size_t)colnc * 3;
        __builtin_prefetch(yn + 48, 0, 1);     // tile ct+2 -> global_prefetch_b8
        float n0 = yn[0], n1 = yn[1], n2 = yn[2];

        // consume current tile
        float ny = cy0 * cy0 + cy1 * cy1 + cy2 * cy2;
        v2f bfrag;
        bfrag.x = half ? cy2 : cy0;
        bfrag.y = half ? 0.f : cy1;

        v8f acc = {0.f, 0.f, 0.f, 0.f, 0.f, 0.f, 0.f, 0.f};
        acc = __builtin_amdgcn_wmma_f32_16x16x4_f32(
            false, afrag, false, bfrag, (short)0, acc, false, false);

#pragma unroll
        for (int r = 0; r < 8; ++r) {
            float d2 = fmaxf(nx[r] + ny - 2.f * acc[r], 0.f);
            rm[r] = fminf(rm[r], d2);
        }

        cy0 = n0; cy1 = n1; cy2 = n2;          // rotate buffers
    }

    // min across the 16 lanes of each half (xor offsets < 16 stay within a half)
#pragma unroll
    for (int r = 0; r < 8; ++r) {
#pragma unroll
        for (int off = 1; off < 16; off <<= 1)
            rm[r] = fminf(rm[r], __shfl_xor(rm[r], off, 32));
    }

    if (m16 == 0) {                            // lane 0 -> rows 0..7, lane 16 -> rows 8..15
#pragma unroll
        for (int r = 0; r < 8; ++r) {
            int m = rt * 16 + r + 8 * half;
            if (m < NN) rowmin[(size_t)b * NN + m] = sqrtf(rm[r]);
        }
    }
}

// ---------------------------------------------------------------------------
// Kernel 2: generalized winding number (Van Oosterom–Strackee), LDS-staged
// gathered triangle soup, one thread per query point. Fast hardware sqrt
// (v_sqrt_f32 via __builtin_amdgcn_sqrtf): the norms only feed an atan2
// argument, so the correctly-rounded refinement sequence is unnecessary.
// ---------------------------------------------------------------------------
__global__ void __launch_bounds__(256)
winding_kernel(const float* __restrict__ pts, const float* __restrict__ verts,
               const int* __restrict__ faces, float* __restrict__ interior)
{
    __shared__ float sA[CHUNK][3], sB[CHUNK][3], sC[CHUNK][3];
    const int tid = threadIdx.x;
    const int nb  = (NN + 255) / 256;
    const int b   = blockIdx.x / nb;
    const int pi  = (blockIdx.x % nb) * 256 + tid;
    const int pic = pi < NN ? pi : NN - 1;

    const float px = pts[((size_t)b * NN + pic) * 3 + 0];
    const float py = pts[((size_t)b * NN + pic) * 3 + 1];
    const float pz = pts[((size_t)b * NN + pic) * 3 + 2];

    float omega = 0.f;
    for (int base = 0; base < NFACES; base += CHUNK) {
        int f = base + tid;
        if (f < NFACES) {                      // gather t = verts[faces] into LDS
            int i0 = faces[f * 3 + 0], i1 = faces[f * 3 + 1], i2 = faces[f * 3 + 2];
            const float* w0 = verts + ((size_t)b * NN + i0) * 3;
            const float* w1 = verts + ((size_t)b * NN + i1) * 3;
            const float* w2 = verts + ((size_t)b * NN + i2) * 3;
            sA[tid][0] = w0[0]; sA[tid][1] = w0[1]; sA[tid][2] = w0[2];
            sB[tid][0] = w1[0]; sB[tid][1] = w1[1]; sB[tid][2] = w1[2];
            sC[tid][0] = w2[0]; sC[tid][1] = w2[1]; sC[tid][2] = w2[2];
        }
        __syncthreads();
        int cnt = NFACES - base; if (cnt > CHUNK) cnt = CHUNK;
        for (int k = 0; k < cnt; ++k) {        // LDS broadcast reads
            float ax = sA[k][0] - px, ay = sA[k][1] - py, az = sA[k][2] - pz;
            float bx = sB[k][0] - px, by = sB[k][1] - py, bz = sB[k][2] - pz;
            float cx = sC[k][0] - px, cy = sC[k][1] - py, cz = sC[k][2] - pz;
            float la = __builtin_amdgcn_sqrtf(ax * ax + ay * ay + az * az);
            float lb = __builtin_amdgcn_sqrtf(bx * bx + by * by + bz * bz);
            float lc = __builtin_amdgcn_sqrtf(cx * cx + cy * cy + cz * cz);
            float ux = by * cz - bz * cy;      // cross(b, c)
            float uy = bz * cx - bx * cz;
            float uz = bx * cy - by * cx;
            float det = ax * ux + ay * uy + az * uz;
            float ab  = ax * bx + ay * by + az * bz;
            float bc  = bx * cx + by * cy + bz * cz;
            float ca  = cx * ax + cy * ay + cz * az;
            float den = la * lb * lc + ab * lc + bc * la + ca * lb;
            omega += 2.f * atan2f(det, den);
        }
        __syncthreads();
    }
    if (pi < NN)
        interior[(size_t)b * NN + pi] = (omega / FOURPI >= 0.99f) ? 1.f : 0.f;
}

// ---------------------------------------------------------------------------
// Kernel 3: masked max/mean + global min reduction, one block per batch.
// ---------------------------------------------------------------------------
__global__ void __launch_bounds__(256)
stats_kernel(const float* __restrict__ rm1, const float* __restrict__ rm2,
             const float* __restrict__ in1, const float* __restrict__ in2,
             float* __restrict__ out)
{
    const int b = blockIdx.x, tid = threadIdx.x;
    float mn = BIGF, mx12 = -BIGF, s12 = 0.f, c12 = 0.f, mx21 = -BIGF, s21 = 0.f, c21 = 0.f;
    for (int i = tid; i < NN; i += 256) {
        float r1 = rm1[(size_t)b * NN + i], r2 = rm2[(size_t)b * NN + i];
        mn = fminf(mn, fminf(r1, r2));
        if (in1[(size_t)b * NN + i] != 0.f) { mx12 = fmaxf(mx12, r1); s12 += r1; c12 += 1.f; }
        if (in2[(size_t)b * NN + i] != 0.f) { mx21 = fmaxf(mx21, r2); s21 += r2; c21 += 1.f; }
    }
    __shared__ float s[7][256];
    s[0][tid] = mn;  s[1][tid] = mx12; s[2][tid] = s12; s[3][tid] = c12;
    s[4][tid] = mx21; s[5][tid] = s21; s[6][tid] = c21;
    __syncthreads();
    for (int off = 128; off > 0; off >>= 1) {
        if (tid < off) {
            s[0][tid] = fminf(s[0][tid], s[0][tid + off]);
            s[1][tid] = fmaxf(s[1][tid], s[1][tid + off]);
            s[2][tid] += s[2][tid + off];
            s[3][tid] += s[3][tid + off];
            s[4][tid] = fmaxf(s[4][tid], s[4][tid + off]);
            s[5][tid] += s[5][tid + off];
            s[6][tid] += s[6][tid + off];
        }
        __syncthreads();
    }
    if (tid == 0) {
        bool a1 = s[3][0] > 0.f, a2 = s[6][0] > 0.f;
        out[0 * BB + b] = s[0][0];
        out[1 * BB + b] = a1 ? s[1][0] : 0.f;
        out[2 * BB + b] = a1 ? s[2][0] / fmaxf(s[3][0], 1.f) : 0.f;
        out[3 * BB + b] = a2 ? s[4][0] : 0.f;
        out[4 * BB + b] = a2 ? s[5][0] / fmaxf(s[6][0], 1.f) : 0.f;
    }
}

extern "C" void kernel_launch(void* const* d_in, const int* in_sizes, int n_in,
                              void* d_out, int out_size, void* d_ws, size_t ws_size,
                              hipStream_t stream)
{
    const float* v1    = (const float*)d_in[0];   // [B,N,3] f32
    const float* v2    = (const float*)d_in[1];   // [B,N,3] f32
    const int*   faces = (const int*)d_in[2];     // [NF,3] i32
    float* out = (float*)d_out;                   // [5,B] f32

    float* rm1 = (float*)d_ws;                    // row_min  [B,N]
    float* rm2 = rm1 + (size_t)BB * NN;           // col_min  [B,N]
    float* in1 = rm2 + (size_t)BB * NN;           // interior_v1 mask
    float* in2 = in1 + (size_t)BB * NN;           // interior_v2 mask

    const int waves  = BB * NTILES;               // 862 waves, 8 waves/block
    const int blocks = (waves + 7) / 8;
    rowmin_wmma_kernel<<<blocks, 256, 0, stream>>>(v1, v2, rm1); // min over v2 per v1 pt
    rowmin_wmma_kernel<<<blocks, 256, 0, stream>>>(v2, v1, rm2); // min over v1 per v2 pt

    const int nb = (NN + 255) / 256;
    winding_kernel<<<BB * nb, 256, 0, stream>>>(v1, v2, faces, in1); // wn(v1, t2)
    winding_kernel<<<BB * nb, 256, 0, stream>>>(v2, v1, faces, in2); // wn(v2, t1)

    stats_kernel<<<BB, 256, 0, stream>>>(rm1, rm2, in1, in2, out);
}